// GCN_48498770706497
// MI455X (gfx1250) — compile-verified
//
#include <hip/hip_runtime.h>
#include <hip/hip_bf16.h>

#define NN   50000
#define FF   128
#define HH   256
#define EE   800000
#define GG   256
#define EPS  1e-5f

typedef float v2f __attribute__((ext_vector_type(2)));
typedef float v8f __attribute__((ext_vector_type(8)));

// ---------------------------------------------------------------------------
// degree: deg[col[e]] += 1   (deg buffer pre-zeroed; +1 self loop in dinv)
// ---------------------------------------------------------------------------
__global__ void k_degree(const int* __restrict__ col, float* __restrict__ deg, int E) {
  int e = blockIdx.x * blockDim.x + threadIdx.x;
  if (e < E) atomicAdd(&deg[col[e]], 1.0f);
}

__global__ void k_dinv(float* __restrict__ deg, int n) {
  int i = blockIdx.x * blockDim.x + threadIdx.x;
  if (i < n) deg[i] = rsqrtf(deg[i] + 1.0f);
}

// ---------------------------------------------------------------------------
// GEMM via V_WMMA_F32_16X16X4_F32.  C[M,NC] = A[M,K] @ B[K,NC], row-major.
// One wave computes a 16x64 strip (4 accumulators); A fragment reused 4x.
// blockIdx.x -> group of 8 16-row tiles (8 waves/block), blockIdx.y -> 64-col
// group.  K/NC are compile-time so the inner loop is pure loads + WMMA with
// pointer increments (no 64-bit muls in the loop).
// Per-lane fragment layout (ISA 7.12.2):
//   A: lane l -> row M=l%16, v0/v1 = K = kbase+(l/16)*2 + {0,1}
//   B: lane l -> col N=l%16, v0/v1 = K = kbase+(l/16)*2 + {0,1}
//   D: vgpr v, lane l -> row v+(l/16)*8, col l%16
// ---------------------------------------------------------------------------
template <int K, int NC>
__global__ __launch_bounds__(256) void k_gemm_wmma(const float* __restrict__ A,
                                                   const float* __restrict__ B,
                                                   float* __restrict__ C, int Mtiles) {
  const int lane = threadIdx.x & 31;
  const int wave = threadIdx.x >> 5;
  const int tm = blockIdx.x * 8 + wave;          // 16-row tile index
  if (tm >= Mtiles) return;                      // wave-uniform: EXEC stays full
  const int cg = blockIdx.y * 64;                // 64-column group base
  const int r  = lane & 15;
  const int kh = (lane >> 4) * 2;                // 0 or 2

  const float* __restrict__ Ap = A + (long)(tm * 16 + r) * K + kh;
  const float* __restrict__ Bp = B + (long)kh * NC + cg + r;

  v8f acc0 = {}, acc1 = {}, acc2 = {}, acc3 = {};
  for (int k = 0; k < K; k += 16) {
    __builtin_prefetch(Ap + 64, 0, 1);
#pragma unroll
    for (int kk = 0; kk < 4; ++kk) {
      const v2f a = *(const v2f*)Ap;
      v2f b0, b1, b2, b3;
      b0.x = Bp[0];       b0.y = Bp[NC];
      b1.x = Bp[16];      b1.y = Bp[NC + 16];
      b2.x = Bp[32];      b2.y = Bp[NC + 32];
      b3.x = Bp[48];      b3.y = Bp[NC + 48];
      acc0 = __builtin_amdgcn_wmma_f32_16x16x4_f32(false, a, false, b0, (short)0, acc0, false, false);
      acc1 = __builtin_amdgcn_wmma_f32_16x16x4_f32(false, a, false, b1, (short)0, acc1, false, false);
      acc2 = __builtin_amdgcn_wmma_f32_16x16x4_f32(false, a, false, b2, (short)0, acc2, false, false);
      acc3 = __builtin_amdgcn_wmma_f32_16x16x4_f32(false, a, false, b3, (short)0, acc3, false, false);
      Ap += 4;
      Bp += 4 * NC;
    }
  }

  const long rowBase = (long)tm * 16 + (lane >> 4) * 8;
  float* __restrict__ Cp = C + rowBase * NC + cg + r;
#pragma unroll
  for (int v = 0; v < 8; ++v) {
    Cp[(long)v * NC +  0] = acc0[v];
    Cp[(long)v * NC + 16] = acc1[v];
    Cp[(long)v * NC + 32] = acc2[v];
    Cp[(long)v * NC + 48] = acc3[v];
  }
}

// ---------------------------------------------------------------------------
// Edge scatter: agg[col] += xw[row] * dinv[row]*dinv[col].  One block per edge,
// 256 threads = 256 features (coalesced load + coalesced atomic add into L2).
// ---------------------------------------------------------------------------
__global__ void k_edge_agg(const float* __restrict__ xw, const int* __restrict__ row,
                           const int* __restrict__ col, const float* __restrict__ dinv,
                           float* __restrict__ agg, int Hd, int E) {
  for (int e = blockIdx.x; e < E; e += gridDim.x) {
    const int r = row[e], c = col[e];
    const float nrm = dinv[r] * dinv[c];
    const float* __restrict__ src = xw + (long)r * Hd;
    float* __restrict__ dst = agg + (long)c * Hd;
    for (int f = threadIdx.x; f < Hd; f += blockDim.x)
      atomicAdd(&dst[f], src[f] * nrm);
  }
}

// ---------------------------------------------------------------------------
// pre = agg + xw*dinv^2 + bias (in-place into agg), accumulate per-feature
// sum / sumsq for BN.  threadIdx.x == feature (256 threads).
// ---------------------------------------------------------------------------
__global__ void k_combine_stats(float* __restrict__ agg, const float* __restrict__ xw,
                                const float* __restrict__ dinv, const float* __restrict__ bias,
                                float* __restrict__ fsum, float* __restrict__ fsq,
                                int n, int Hd, int rowsPerBlock) {
  const int f  = threadIdx.x;
  const int i0 = blockIdx.x * rowsPerBlock;
  const int i1 = min(i0 + rowsPerBlock, n);
  const float b = bias[f];
  float s = 0.f, ss = 0.f;
  for (int i = i0; i < i1; ++i) {
    const float d = dinv[i];
    const long  o = (long)i * Hd + f;
    const float v = agg[o] + xw[o] * d * d + b;
    agg[o] = v;
    s += v; ss += v * v;
  }
  atomicAdd(&fsum[f], s);
  atomicAdd(&fsq[f], ss);
}

// ---------------------------------------------------------------------------
// h = [h +] relu(g*(pre-mu)*rsqrt(var+eps)+beta)   (in-place residual)
// ---------------------------------------------------------------------------
__global__ void k_bn_relu_res(const float* __restrict__ pre, const float* __restrict__ fsum,
                              const float* __restrict__ fsq, const float* __restrict__ g,
                              const float* __restrict__ beta, float* __restrict__ h,
                              int n, int Hd, int residual, int rowsPerBlock) {
  const int f  = threadIdx.x;
  const float invN = 1.0f / (float)n;
  const float mu   = fsum[f] * invN;
  const float var  = fsq[f] * invN - mu * mu;
  const float sc   = g[f] * rsqrtf(var + EPS);
  const float bb   = beta[f] - mu * sc;
  const int i0 = blockIdx.x * rowsPerBlock;
  const int i1 = min(i0 + rowsPerBlock, n);
  for (int i = i0; i < i1; ++i) {
    const long o = (long)i * Hd + f;
    const float v = fmaxf(pre[o] * sc + bb, 0.0f);
    h[o] = residual ? (h[o] + v) : v;
  }
}

// ---------------------------------------------------------------------------
// Pooling: counts + per-group feature sums (batch_ids sorted -> run-length
// compress before hitting atomics).
// ---------------------------------------------------------------------------
__global__ void k_counts(const int* __restrict__ batch, float* __restrict__ counts, int n) {
  int i = blockIdx.x * blockDim.x + threadIdx.x;
  if (i < n) atomicAdd(&counts[batch[i]], 1.0f);
}

__global__ void k_pool_sums(const float* __restrict__ h, const int* __restrict__ batch,
                            float* __restrict__ sums, int n, int Hd, int rowsPerBlock) {
  const int f  = threadIdx.x;
  const int i0 = blockIdx.x * rowsPerBlock;
  const int i1 = min(i0 + rowsPerBlock, n);
  float acc = 0.f;
  int cur = -1;
  for (int i = i0; i < i1; ++i) {
    const int b = batch[i];
    if (b != cur) {
      if (cur >= 0) atomicAdd(&sums[(long)cur * Hd + f], acc);
      cur = b; acc = 0.f;
    }
    acc += h[(long)i * Hd + f];
  }
  if (cur >= 0) atomicAdd(&sums[(long)cur * Hd + f], acc);
}

// ---------------------------------------------------------------------------
// Per-group MLP head: pooled = sums/max(cnt,1)+sums;
// out[g] = relu(pooled@mW1+mb1)@mW2 + mb2.   128 threads, one block per group.
// ---------------------------------------------------------------------------
__global__ void k_mlp(const float* __restrict__ sums, const float* __restrict__ counts,
                      const float* __restrict__ mW1, const float* __restrict__ mb1,
                      const float* __restrict__ mW2, const float* __restrict__ mb2,
                      float* __restrict__ out, int Hd, int Hh) {
  __shared__ float pooled[HH];
  __shared__ float red[HH / 2];
  const int g = blockIdx.x;
  const int j = threadIdx.x;
  const float cnt = fmaxf(counts[g], 1.0f);
  for (int f = j; f < Hd; f += blockDim.x) {
    const float s = sums[(long)g * Hd + f];
    pooled[f] = s / cnt + s;
  }
  __syncthreads();
  float acc = mb1[j];
  for (int f = 0; f < Hd; ++f)
    acc = fmaf(pooled[f], mW1[(long)f * Hh + j], acc);
  red[j] = fmaxf(acc, 0.0f) * mW2[j];
  __syncthreads();
  for (int s = Hh >> 1; s > 0; s >>= 1) {
    if (j < s) red[j] += red[j + s];
    __syncthreads();
  }
  if (j == 0) out[g] = red[0] + mb2[0];
}

// ---------------------------------------------------------------------------
extern "C" void kernel_launch(void* const* d_in, const int* in_sizes, int n_in,
                              void* d_out, int out_size, void* d_ws, size_t ws_size,
                              hipStream_t stream) {
  const float* x     = (const float*)d_in[0];
  const int*   edges = (const int*)d_in[1];       // [2,E] flat: row then col
  const int*   batch = (const int*)d_in[2];
  const float* W[3]  = { (const float*)d_in[3],  (const float*)d_in[7],  (const float*)d_in[11] };
  const float* bi[3] = { (const float*)d_in[4],  (const float*)d_in[8],  (const float*)d_in[12] };
  const float* gm[3] = { (const float*)d_in[5],  (const float*)d_in[9],  (const float*)d_in[13] };
  const float* bt[3] = { (const float*)d_in[6],  (const float*)d_in[10], (const float*)d_in[14] };
  const float* mW1 = (const float*)d_in[15];
  const float* mb1 = (const float*)d_in[16];
  const float* mW2 = (const float*)d_in[17];
  const float* mb2 = (const float*)d_in[18];
  float* out = (float*)d_out;

  const int* rowI = edges;
  const int* colI = edges + EE;

  // workspace carve-up (floats)
  float* ws = (float*)d_ws;
  size_t off = 0;
  float* dinv = ws + off; off += 50176;                    // N padded
  float* xw   = ws + off; off += (size_t)NN * HH;
  float* agg  = ws + off; off += (size_t)NN * HH;          // becomes "pre"
  float* h    = ws + off; off += (size_t)NN * HH;
  float* fsum = ws + off; off += 256;
  float* fsq  = ws + off; off += 256;
  float* cnts = ws + off; off += 256;
  float* gsum = ws + off; off += (size_t)GG * HH;

  const int RPB = 64;                                      // rows per block
  const int rowBlocks = (NN + RPB - 1) / RPB;

  const int Mtiles = NN / 16;                              // 3125
  const dim3 gemmGrid((Mtiles + 7) / 8, HH / 64);          // 391 x 4

  // degree -> dinv
  hipMemsetAsync(dinv, 0, (size_t)NN * sizeof(float), stream);
  k_degree<<<(EE + 255) / 256, 256, 0, stream>>>(colI, dinv, EE);
  k_dinv<<<(NN + 255) / 256, 256, 0, stream>>>(dinv, NN);

  for (int L = 0; L < 3; ++L) {
    if (L == 0)
      k_gemm_wmma<FF, HH><<<gemmGrid, 256, 0, stream>>>(x, W[0], xw, Mtiles);
    else
      k_gemm_wmma<HH, HH><<<gemmGrid, 256, 0, stream>>>(h, W[L], xw, Mtiles);

    hipMemsetAsync(agg,  0, (size_t)NN * HH * sizeof(float), stream);
    hipMemsetAsync(fsum, 0, 256 * sizeof(float), stream);
    hipMemsetAsync(fsq,  0, 256 * sizeof(float), stream);

    k_edge_agg<<<EE, 256, 0, stream>>>(xw, rowI, colI, dinv, agg, HH, EE);
    k_combine_stats<<<rowBlocks, 256, 0, stream>>>(agg, xw, dinv, bi[L], fsum, fsq, NN, HH, RPB);
    k_bn_relu_res<<<rowBlocks, 256, 0, stream>>>(agg, fsum, fsq, gm[L], bt[L], h,
                                                 NN, HH, (L > 0) ? 1 : 0, RPB);
  }

  // pooling + head
  hipMemsetAsync(cnts, 0, 256 * sizeof(float), stream);
  hipMemsetAsync(gsum, 0, (size_t)GG * HH * sizeof(float), stream);
  k_counts<<<(NN + 255) / 256, 256, 0, stream>>>(batch, cnts, NN);
  k_pool_sums<<<rowBlocks, 256, 0, stream>>>(h, batch, gsum, NN, HH, RPB);
  k_mlp<<<GG, HH / 2, 0, stream>>>(gsum, cnts, mW1, mb1, mW2, mb2, out, HH, HH / 2);
}